// GAT_32530082300265
// MI455X (gfx1250) — compile-verified
//
#include <hip/hip_runtime.h>
#include <hip/hip_bf16.h>

// ---------------------------------------------------------------------------
// GAT (3-layer, PyG GATConv style) for gfx1250 / MI455X.
// fp32 WMMA (v_wmma_f32_16x16x4_f32) for the node-feature GEMMs, LDS-staged
// A and B tiles (float4 global loads), hardware fp32 global atomics for the
// segment softmax / aggregation scatters.
// ---------------------------------------------------------------------------

typedef __attribute__((ext_vector_type(2))) float v2f;
typedef __attribute__((ext_vector_type(8))) float v8f;

#define GAT_H   4
#define GAT_NEG 0.2f

#define WPB  8              // waves per block
#define TM   (16 * WPB)     // 128 rows of C per block
#define KC   32             // K chunk staged in LDS

__device__ __forceinline__ void atomicAddF(float* p, float v) {
    __hip_atomic_fetch_add(p, v, __ATOMIC_RELAXED, __HIP_MEMORY_SCOPE_AGENT);
}

// monotone float -> uint key (order preserving), for atomicMax on floats
__device__ __forceinline__ unsigned int f2key(float f) {
    unsigned int b = __float_as_uint(f);
    return (b & 0x80000000u) ? ~b : (b | 0x80000000u);
}
__device__ __forceinline__ float key2f(unsigned int k) {
    unsigned int b = (k & 0x80000000u) ? (k & 0x7FFFFFFFu) : ~k;
    return __uint_as_float(b);
}

// ---------------------------------------------------------------------------
// C[M,Nc] = A[M,K] @ B[K,Nc], fp32, WMMA 16x16x4.
// grid = (Nc/16, ceil(M/TM)), block = 256 (8 waves), 1 wave -> one 16x16 tile.
// Requires K % KC == 0 and Nc % 16 == 0 (true for all three layers).
// ---------------------------------------------------------------------------
__global__ __launch_bounds__(256, 2) void gat_gemm_wmma(
    const float* __restrict__ A, const float* __restrict__ B,
    float* __restrict__ C, int M, int K, int Nc)
{
    __shared__ float sA[KC][TM + 1];   // [k][m], padded vs bank conflicts
    __shared__ float sB[KC][16 + 1];   // [k][n], shared by all 8 waves

    const int tid   = threadIdx.x;
    const int lane  = tid & 31;
    const int wave  = tid >> 5;
    const int m0    = blockIdx.y * TM;
    const int n0    = blockIdx.x * 16;
    const int row16 = lane & 15;            // M (A) / N (B,C) index within tile
    const int kpair = (lane >> 4) << 1;     // 0 or 2: which K pair this lane holds
    const bool fullM = (m0 + TM <= M);

    v8f acc = {0.f, 0.f, 0.f, 0.f, 0.f, 0.f, 0.f, 0.f};

    for (int k0 = 0; k0 < K; k0 += KC) {
        // ---- stage A[m0:m0+TM, k0:k0+KC] into sA[k][m] ----
        if (fullM) {
            // 128 rows x 32 k = 1024 float4; 256 threads x 4, fully unrolled
            #pragma unroll
            for (int j = 0; j < 4; ++j) {
                int i  = tid + j * 256;        // [0,1024)
                int m  = i >> 3;               // 8 float4 per row
                int k4 = (i & 7) << 2;
                float4 v = *(const float4*)&A[(size_t)(m0 + m) * K + k0 + k4];
                sA[k4 + 0][m] = v.x;
                sA[k4 + 1][m] = v.y;
                sA[k4 + 2][m] = v.z;
                sA[k4 + 3][m] = v.w;
            }
        } else {
            for (int i = tid; i < TM * KC; i += 256) {
                int m = i >> 5;            // KC == 32
                int k = i & (KC - 1);
                int gm = m0 + m;
                sA[k][m] = (gm < M) ? A[(size_t)gm * K + k0 + k] : 0.f;
            }
        }
        // ---- stage B[k0:k0+KC, n0:n0+16] into sB[k][n] ----
        if (tid < 128) {                       // 32 rows x 4 float4
            int kr = tid >> 2;
            int c4 = (tid & 3) << 2;
            float4 v = *(const float4*)&B[(size_t)(k0 + kr) * Nc + n0 + c4];
            sB[kr][c4 + 0] = v.x;
            sB[kr][c4 + 1] = v.y;
            sB[kr][c4 + 2] = v.z;
            sB[kr][c4 + 3] = v.w;
        }
        __syncthreads();

        const int mloc = wave * 16 + row16;
        #pragma unroll
        for (int kk = 0; kk < KC; kk += 4) {
            v2f a, b;
            a.x = sA[kk + kpair    ][mloc];
            a.y = sA[kk + kpair + 1][mloc];
            b.x = sB[kk + kpair    ][row16];
            b.y = sB[kk + kpair + 1][row16];
            acc = __builtin_amdgcn_wmma_f32_16x16x4_f32(
                false, a, false, b, (short)0, acc, false, false);
        }
        __syncthreads();
    }

    // C/D layout: VGPR r -> rows r (lanes 0-15) and r+8 (lanes 16-31)
    #pragma unroll
    for (int r = 0; r < 8; ++r) {
        int gm = m0 + wave * 16 + r + ((lane >> 4) << 3);
        if (gm < M) C[(size_t)gm * Nc + n0 + row16] = acc[r];
    }
}

// ---------------------------------------------------------------------------
// al_s[n,h] = <h[n,h,:], a_src[h,:]> ; al_d likewise. One wave per (n,h).
// ---------------------------------------------------------------------------
__global__ void gat_attn_coef(const float* __restrict__ h,
                              const float* __restrict__ a_src,
                              const float* __restrict__ a_dst,
                              float* __restrict__ al_s,
                              float* __restrict__ al_d,
                              int Nn, int Cc)
{
    int wid  = blockIdx.x * (blockDim.x >> 5) + (threadIdx.x >> 5);
    int lane = threadIdx.x & 31;
    if (wid >= Nn * GAT_H) return;
    int hd = wid & (GAT_H - 1);
    const float* hp = h + (size_t)wid * Cc;          // h is [N, H, C] contiguous
    const float* as = a_src + hd * Cc;
    const float* ad = a_dst + hd * Cc;
    float ss = 0.f, sd = 0.f;
    for (int c = lane; c < Cc; c += 32) {
        float v = hp[c];
        ss += v * as[c];
        sd += v * ad[c];
    }
    for (int off = 16; off > 0; off >>= 1) {
        ss += __shfl_down(ss, off, 32);
        sd += __shfl_down(sd, off, 32);
    }
    if (lane == 0) { al_s[wid] = ss; al_d[wid] = sd; }
}

__global__ void gat_fill_u32(unsigned int* __restrict__ p, unsigned int v, int n)
{
    int i = blockIdx.x * blockDim.x + threadIdx.x;
    if (i < n) p[i] = v;
}

// ---------------------------------------------------------------------------
// Edge pass 1: logits = leaky_relu(al_s[src]+al_d[dst]); segment max via
// atomicMax on order-preserving uint keys.  Edges e<E from edge_index,
// e>=E are self loops.
// ---------------------------------------------------------------------------
__global__ void gat_edge_logits(const long long* __restrict__ ei,
                                const float* __restrict__ al_s,
                                const float* __restrict__ al_d,
                                float* __restrict__ logits,
                                unsigned int* __restrict__ mxkey,
                                int E, int Nn)
{
    int e = blockIdx.x * blockDim.x + threadIdx.x;
    if (e >= E + Nn) return;
    int s, d;
    if (e < E) { s = (int)ei[e]; d = (int)ei[E + e]; } else { s = d = e - E; }
    #pragma unroll
    for (int hd = 0; hd < GAT_H; ++hd) {
        float l = al_s[s * GAT_H + hd] + al_d[d * GAT_H + hd];
        l = (l > 0.f) ? l : l * GAT_NEG;
        logits[(size_t)e * GAT_H + hd] = l;
        atomicMax(&mxkey[d * GAT_H + hd], f2key(l));
    }
}

// Edge pass 2: ex = exp(logit - mx[dst]); den[dst] += ex (in-place over logits)
__global__ void gat_edge_exp(const long long* __restrict__ ei,
                             float* __restrict__ logits,
                             const unsigned int* __restrict__ mxkey,
                             float* __restrict__ den,
                             int E, int Nn)
{
    int e = blockIdx.x * blockDim.x + threadIdx.x;
    if (e >= E + Nn) return;
    int d = (e < E) ? (int)ei[E + e] : (e - E);
    #pragma unroll
    for (int hd = 0; hd < GAT_H; ++hd) {
        float mx = key2f(mxkey[d * GAT_H + hd]);
        if (!isfinite(mx)) mx = 0.f;
        float ex = __expf(logits[(size_t)e * GAT_H + hd] - mx);
        logits[(size_t)e * GAT_H + hd] = ex;
        atomicAddF(&den[d * GAT_H + hd], ex);
    }
}

// Edge pass 3: agg[dst, h, c] += h[src, h, c] * ex/(den[dst,h]+eps)
// one block per edge, blockDim = H*C (256 or 160)
__global__ void gat_edge_msg(const long long* __restrict__ ei,
                             const float* __restrict__ hfeat,
                             const float* __restrict__ ex,
                             const float* __restrict__ den,
                             float* __restrict__ agg,
                             int E, int Cc)
{
    int e = blockIdx.x;
    int t = threadIdx.x;                 // channel index in [0, H*C)
    int s, d;
    if (e < E) { s = (int)ei[e]; d = (int)ei[E + e]; } else { s = d = e - E; }
    int hd = t / Cc;
    float alpha = ex[(size_t)e * GAT_H + hd] / (den[d * GAT_H + hd] + 1e-16f);
    int HC = GAT_H * Cc;
    atomicAddF(&agg[(size_t)d * HC + t], hfeat[(size_t)s * HC + t] * alpha);
}

// out = elu(agg + bias [+ resid])
__global__ void gat_elu_bias(const float* __restrict__ agg,
                             const float* __restrict__ bias,
                             const float* __restrict__ resid,
                             float* __restrict__ out, int total, int HC)
{
    int i = blockIdx.x * blockDim.x + threadIdx.x;
    if (i >= total) return;
    float v = agg[i] + bias[i & (HC - 1)];   // HC == 256 here
    if (resid) v += resid[i];
    out[i] = (v > 0.f) ? v : (__expf(v) - 1.f);
}

// out[n, j] = mean_h agg[n, h, j] + bias[j]
__global__ void gat_head_mean(const float* __restrict__ agg,
                              const float* __restrict__ bias,
                              float* __restrict__ out, int Nn, int NCc)
{
    int i = blockIdx.x * blockDim.x + threadIdx.x;
    if (i >= Nn * NCc) return;
    int n = i / NCc, j = i - n * NCc;
    float s = 0.f;
    #pragma unroll
    for (int hd = 0; hd < GAT_H; ++hd)
        s += agg[(size_t)n * GAT_H * NCc + hd * NCc + j];
    out[i] = s * (1.f / GAT_H) + bias[j];
}

// ---------------------------------------------------------------------------
extern "C" void kernel_launch(void* const* d_in, const int* in_sizes, int n_in,
                              void* d_out, int out_size, void* d_ws, size_t ws_size,
                              hipStream_t stream)
{
    const float*     x   = (const float*)d_in[0];
    const long long* ei  = (const long long*)d_in[1];
    const float*     W1  = (const float*)d_in[2];
    const float*     a1s = (const float*)d_in[3];
    const float*     a1d = (const float*)d_in[4];
    const float*     b1  = (const float*)d_in[5];
    const float*     W2  = (const float*)d_in[6];
    const float*     a2s = (const float*)d_in[7];
    const float*     a2d = (const float*)d_in[8];
    const float*     b2  = (const float*)d_in[9];
    const float*     W3  = (const float*)d_in[10];
    const float*     a3s = (const float*)d_in[11];
    const float*     a3d = (const float*)d_in[12];
    const float*     b3  = (const float*)d_in[13];

    const int F  = 128, C = 64, NC = 40, HC = GAT_H * C;
    const int Nn = in_sizes[0] / F;
    const int E  = in_sizes[1] / 2;
    const int ET = E + Nn;

    // workspace layout (floats)
    float* ws   = (float*)d_ws;
    float* h    = ws;                       // N * 256
    float* agg  = h   + (size_t)Nn * HC;    // N * 256
    float* x1   = agg + (size_t)Nn * HC;    // N * 256
    float* x2   = x1  + (size_t)Nn * HC;    // N * 256
    float* als  = x2  + (size_t)Nn * HC;    // N * H
    float* ald  = als + (size_t)Nn * GAT_H; // N * H
    unsigned int* mx = (unsigned int*)(ald + (size_t)Nn * GAT_H); // N * H
    float* den  = (float*)(mx + (size_t)Nn * GAT_H);              // N * H
    float* exb  = den + (size_t)Nn * GAT_H;                       // ET * H

    const unsigned int NEG_INF_KEY = 0x007FFFFFu;  // f2key(-inf)

    auto run_layer = [&](const float* X, int K, const float* W,
                         const float* as, const float* ad, int Cc) {
        const int HCl = GAT_H * Cc;
        // init per-layer accumulators
        gat_fill_u32<<<(Nn * GAT_H + 255) / 256, 256, 0, stream>>>(mx, NEG_INF_KEY, Nn * GAT_H);
        gat_fill_u32<<<(Nn * GAT_H + 255) / 256, 256, 0, stream>>>((unsigned int*)den, 0u, Nn * GAT_H);
        gat_fill_u32<<<(Nn * HCl + 255) / 256, 256, 0, stream>>>((unsigned int*)agg, 0u, Nn * HCl);
        // h = X @ W   (fp32 WMMA)
        dim3 gg(HCl / 16, (Nn + TM - 1) / TM);
        gat_gemm_wmma<<<gg, 256, 0, stream>>>(X, W, h, Nn, K, HCl);
        // attention coefficients
        gat_attn_coef<<<(Nn * GAT_H + WPB - 1) / WPB, 256, 0, stream>>>(h, as, ad, als, ald, Nn, Cc);
        // edge passes
        gat_edge_logits<<<(ET + 255) / 256, 256, 0, stream>>>(ei, als, ald, exb, mx, E, Nn);
        gat_edge_exp   <<<(ET + 255) / 256, 256, 0, stream>>>(ei, exb, mx, den, E, Nn);
        gat_edge_msg   <<<ET, HCl, 0, stream>>>(ei, h, exb, den, agg, E, Cc);
    };

    // layer 1: x1 = elu(gat(x) + b1)
    run_layer(x, F, W1, a1s, a1d, C);
    gat_elu_bias<<<((Nn * HC) + 255) / 256, 256, 0, stream>>>(agg, b1, nullptr, x1, Nn * HC, HC);
    // layer 2: x2 = elu(gat(x1) + b2 + x1)
    run_layer(x1, HC, W2, a2s, a2d, C);
    gat_elu_bias<<<((Nn * HC) + 255) / 256, 256, 0, stream>>>(agg, b2, x1, x2, Nn * HC, HC);
    // layer 3: out = mean_heads(gat(x2)) + b3
    run_layer(x2, HC, W3, a3s, a3d, NC);
    gat_head_mean<<<((Nn * NC) + 255) / 256, 256, 0, stream>>>(agg, b3, (float*)d_out, Nn, NC);
    (void)n_in; (void)out_size; (void)ws_size;
}